// PhysicsInformedLoss_21191368638890
// MI455X (gfx1250) — compile-verified
//
#include <hip/hip_runtime.h>
#include <cstdint>

#define SLEN 8192
#define BDIM 256
#define NQ 12

typedef __attribute__((ext_vector_type(2))) float v2f;
typedef __attribute__((ext_vector_type(8))) float v8f;

__device__ __forceinline__ unsigned lds_off(const void* p) {
  // AS3 -> generic flat addr: low 32 bits are the LDS byte offset (aperture in [63:32])
  return (unsigned)(uintptr_t)p;
}

__device__ __forceinline__ void async_ld_b128(unsigned ldsaddr, const void* g) {
  asm volatile("global_load_async_to_lds_b128 %0, %1, off"
               :: "v"(ldsaddr), "v"(g) : "memory");
}

__device__ __forceinline__ void wait_async0() {
  asm volatile("s_wait_asynccnt 0" ::: "memory");
}

__global__ __launch_bounds__(BDIM) void pil_row_kernel(
    const float* __restrict__ pred, const float* __restrict__ tgt,
    const float* __restrict__ bp, const unsigned char* __restrict__ mask,
    const float* __restrict__ voltage, const float* __restrict__ thick,
    float* __restrict__ partial /* [NQ][B] */, int B)
{
  __shared__ float s_pred[SLEN];
  __shared__ unsigned char s_mask[SLEN];
  __shared__ float s_red[BDIM / 32][11];
  __shared__ int s_last[BDIM / 32];

  const int row = blockIdx.x;
  const int tid = threadIdx.x;
  const size_t base = (size_t)row * SLEN;

  const unsigned p0 = lds_off(&s_pred[0]);
  const unsigned m0 = lds_off(&s_mask[0]);

  // Async-stage the row: predictions 32KB (8 x b128/lane), mask 8KB (2 x b128/lane).
  // Each instruction moves a contiguous 512B per wave.
#pragma unroll
  for (int k = 0; k < 8; ++k) {
    const int e = k * 1024 + tid * 4;            // float index
    async_ld_b128(p0 + (unsigned)(e * 4), pred + base + e);
  }
#pragma unroll
  for (int k = 0; k < 2; ++k) {
    const int e = k * 4096 + tid * 16;           // byte index
    async_ld_b128(m0 + (unsigned)e, mask + base + e);
  }
  wait_async0();
  __syncthreads();

  float a_msum = 0.f, a_se = 0.f, a_bce = 0.f, a_mono = 0.f, a_vps = 0.f,
        a_smooth = 0.f, a_vts = 0.f, a_bnd = 0.f, a_vd = 0.f, a_cnt = 0.f,
        a_any1 = 0.f;
  int last = -1;

  for (int it = 0; it < SLEN / BDIM; ++it) {
    const int j = tid + it * BDIM;               // coalesced global, conflict-free LDS
    const float m = (float)s_mask[j];
    const float pj = s_pred[j];
    const float t = tgt[base + j];
    const float q = bp[base + j];

    a_msum += m;
    const float e = pj - t;
    a_se += e * e * m;
    a_bnd += fmaxf(pj - 200.0f, 0.0f);
    const float pc = fminf(fmaxf(q, 1e-7f), 1.0f - 1e-7f);
    a_bce += (-log1pf(-pc)) * m;                 // base BCE (bt==0), weight = m
    if (m != 0.0f) last = j;
    if (j < SLEN - 1) a_cnt += m;
    if (j >= 1) {
      const float m1 = (float)s_mask[j - 1];
      const float p1 = s_pred[j - 1];
      const float d = pj - p1;
      const float pm = m * m1;
      a_mono += fmaxf(-d, 0.0f) * pm;
      a_vps += pm;
      a_vd += d * m1;                            // (pred[j]-pred[j-1]) * m[j-1]
      a_any1 += m;
      if (j >= 2) {
        const float m2 = (float)s_mask[j - 2];
        const float p2 = s_pred[j - 2];
        const float sd = pj - 2.0f * p1 + p2;
        const float tm = pm * m2;
        a_smooth += fabsf(sd) * tm;
        a_vts += tm;
      }
    }
  }

  float acc[11] = {a_msum, a_se, a_bce, a_mono, a_vps, a_smooth,
                   a_vts, a_bnd, a_vd, a_cnt, a_any1};
#pragma unroll
  for (int off = 16; off > 0; off >>= 1) {
#pragma unroll
    for (int k = 0; k < 11; ++k) acc[k] += __shfl_down(acc[k], off, 32);
    const int o = __shfl_down(last, off, 32);
    last = last > o ? last : o;
  }
  const int wave = tid >> 5;
  if ((tid & 31) == 0) {
#pragma unroll
    for (int k = 0; k < 11; ++k) s_red[wave][k] = acc[k];
    s_last[wave] = last;
  }
  __syncthreads();

  if (tid == 0) {
    float r[11];
    for (int k = 0; k < 11; ++k) r[k] = s_red[0][k];
    int L = s_last[0];
    for (int w = 1; w < BDIM / 32; ++w) {
      for (int k = 0; k < 11; ++k) r[k] += s_red[w][k];
      L = L > s_last[w] ? L : s_last[w];
    }
    // breakdown-target correction at the last valid index
    float bce_w = r[2];
    float wsum = r[0];
    if (L >= 0) {
      const float tl = tgt[base + L];
      const bool cond = (L < SLEN - 1) || (tl >= 190.0f);  // THR*0.95
      if (cond) {
        const float q = bp[base + L];
        const float pc = fminf(fmaxf(q, 1e-7f), 1.0f - 1e-7f);
        bce_w += 5.0f * (-logf(pc)) - (-log1pf(-pc));
        wsum += 4.0f;                                      // weight 1 -> 5
      }
    }
    // physics per-row term
    float phys = 0.0f, npos = 0.0f;
    const float avg = r[8] / fmaxf(r[9], 1.0f);
    if (avg > 0.0f) {
      const float ef = fabsf(voltage[row]) / thick[row] * 1e-7f;
      const float ff = expf(0.1f * ef);
      const float lr = logf(fmaxf(avg, 1e-8f));
      const float lff = logf(fmaxf(ff, 1e-8f));
      const float dd = lr - lff;
      phys = dd * dd;
      npos = 1.0f;
    }
    const float out12[NQ] = {r[0], r[1], bce_w, wsum, r[3], r[4],
                             r[5], r[6], r[7], phys, npos, r[10]};
    for (int q = 0; q < NQ; ++q) partial[q * B + row] = out12[q];
  }
}

// Single-wave finalize: reduce 4096 row-partials per quantity with
// v_wmma_f32_16x16x4_f32 using an all-ones A matrix (D[0][n] = sum_k B[k][n]
// accumulated into C), then horizontally sum the 16 column sums.
__global__ __launch_bounds__(32) void pil_finalize_kernel(
    const float* __restrict__ partial, float* __restrict__ out, int B)
{
  __shared__ float totals[NQ];
  const int lane = threadIdx.x;
  v2f a;
  a.x = 1.0f;
  a.y = 1.0f;
  const int sel = (lane < 16) ? 0 : 2;
  const int n = lane & 15;

  for (int q = 0; q < NQ; ++q) {
    v8f c = {};
    const float* d = partial + q * B;
    for (int base = 0; base < B; base += 64) {  // B % 64 == 0 (B = 4096)
      v2f b;
      b.x = d[base + sel * 16 + n];
      b.y = d[base + (sel + 1) * 16 + n];
      c = __builtin_amdgcn_wmma_f32_16x16x4_f32(
          /*neg_a=*/false, a, /*neg_b=*/false, b,
          /*c_mod=*/(short)0, c, /*reuse_a=*/false, /*reuse_b=*/false);
    }
    float v = c[0];                             // row 0 column sums (lanes 0..15)
#pragma unroll
    for (int off = 8; off > 0; off >>= 1) v += __shfl_down(v, off, 32);
    if (lane == 0) totals[q] = v;
  }

  if (lane == 0) {
    const float msum = totals[0], se = totals[1], bce = totals[2],
                wsum = totals[3], mono = totals[4], vps = totals[5],
                smooth = totals[6], vts = totals[7], bnd = totals[8],
                phys = totals[9], npos = totals[10], any1 = totals[11];
    const bool any_valid = msum > 0.0f;
    const float pred_loss = any_valid ? se / fmaxf(msum, 1.0f) : 0.0f;
    const float bd_loss = any_valid ? bce / fmaxf(wsum, 1.0f) : 0.0f;
    const float mono_loss = (vps > 0.0f) ? mono / fmaxf(vps, 1.0f) : 0.0f;
    const float smooth_loss = (vts > 0.0f) ? smooth / fmaxf(vts, 1.0f) : 0.0f;
    const float phys_mse = phys / fmaxf(npos, 1.0f);
    const float phys_loss = (any1 > 0.0f && npos > 0.0f) ? phys_mse : 0.0f;
    const float bnd_loss = bnd / ((float)B * (float)SLEN) * 0.1f;
    const float total = 1.0f * pred_loss + 0.5f * bd_loss + 0.2f * mono_loss +
                        0.1f * smooth_loss + 0.3f * phys_loss + bnd_loss;
    out[0] = pred_loss;
    out[1] = bd_loss;
    out[2] = mono_loss;
    out[3] = smooth_loss;
    out[4] = phys_loss;
    out[5] = bnd_loss;
    out[6] = total;
  }
}

extern "C" void kernel_launch(void* const* d_in, const int* in_sizes, int n_in,
                              void* d_out, int out_size, void* d_ws, size_t ws_size,
                              hipStream_t stream) {
  const float* pred = (const float*)d_in[0];
  const float* tgt = (const float*)d_in[1];
  const float* bp = (const float*)d_in[2];
  const unsigned char* mask = (const unsigned char*)d_in[3];  // bool -> 1 byte
  const float* volt = (const float*)d_in[4];
  const float* thick = (const float*)d_in[5];
  const int B = in_sizes[4];  // voltage length = 4096 rows

  float* partial = (float*)d_ws;  // NQ * B floats = 192 KB

  pil_row_kernel<<<B, BDIM, 0, stream>>>(pred, tgt, bp, mask, volt, thick,
                                         partial, B);
  pil_finalize_kernel<<<1, 32, 0, stream>>>(partial, (float*)d_out, B);
}